// QuantizeContents_12790412607538
// MI455X (gfx1250) — compile-verified
//
#include <hip/hip_runtime.h>

typedef __attribute__((ext_vector_type(2))) float v2f;
typedef __attribute__((ext_vector_type(8))) float v8f;

#define KDIM 80
#define LDS_STRIDE 84   // 80 + 4 pad: avoid bank conflicts on column access

// ---------------- kernel 0: zero the loss accumulator ----------------
__global__ void vq_init_kernel(float* acc) { acc[0] = 0.0f; }

// ---------------- kernel 1: inverse norms of embedding rows ----------------
__global__ void vq_invnorm_kernel(const float* __restrict__ W, float* __restrict__ invn, int E) {
    int j = blockIdx.x * blockDim.x + threadIdx.x;
    if (j < E) {
        const float* r = W + (size_t)j * KDIM;
        float s = 0.0f;
#pragma unroll
        for (int k = 0; k < KDIM; ++k) { float w = r[k]; s += w * w; }
        invn[j] = 1.0f / sqrtf(s);
    }
}

// ---------------- kernel 2: fused GEMM (X * W^T) + scaled argmax ----------------
// One wave32 per 16-row tile; 8 waves/block share double-buffered B tiles in LDS.
// A frag (16x4 f32): lane l -> M=l%16, v.x = A[M][kb+(l<16?0:2)], v.y = +1
// B frag (4x16 f32): lane l -> N=l%16, v.x = B[kb+(l<16?0:2)][N], v.y = +1
// C/D (16x16 f32):   lane l -> N=l%16, VGPR v -> M = v + (l<16?0:8)
__global__ __launch_bounds__(256) void vq_argmax_kernel(
    const float* __restrict__ X, const float* __restrict__ W,
    const float* __restrict__ invn, int* __restrict__ ind, int E) {
    __shared__ float bt[2][16 * LDS_STRIDE];

    const int t    = threadIdx.x;
    const int lane = t & 31;
    const int wave = t >> 5;
    const int rowTile = blockIdx.x * 8 + wave;
    const int n16  = lane & 15;
    const int koff = (lane < 16) ? 0 : 2;

    // Per-thread staging geometry: 16*80 = 1280 = 5*256 elements per tile.
    int loff[5], goff[5];
#pragma unroll
    for (int it = 0; it < 5; ++it) {
        int p = t + 256 * it;
        int n = p / KDIM;
        int k = p - n * KDIM;
        loff[it] = n * LDS_STRIDE + k;
        goff[it] = n * KDIM + k;
    }

    // Prologue: stage B tile 0.
#pragma unroll
    for (int it = 0; it < 5; ++it) bt[0][loff[it]] = W[goff[it]];

    // A fragments for this wave's 16 rows (held in registers for all 64 tiles).
    v2f a[20];
    const float* arow = X + (size_t)(rowTile * 16 + n16) * KDIM;
#pragma unroll
    for (int kk = 0; kk < 20; ++kk) {
        a[kk].x = arow[4 * kk + koff];
        a[kk].y = arow[4 * kk + koff + 1];
    }

    float best[8];
    int   bidx[8];
#pragma unroll
    for (int v = 0; v < 8; ++v) { best[v] = -__builtin_inff(); bidx[v] = 0; }

    __syncthreads();

    const int nColTiles = E >> 4;
    for (int ct = 0; ct < nColTiles; ++ct) {
        const int cur = ct & 1;

        // Prefetch next tile from global (L2-resident) into registers,
        // overlapping with this tile's WMMA chain.
        float r[5];
        const bool more = (ct + 1) < nColTiles;
        if (more) {
            const float* wt = W + (size_t)(ct + 1) * 16 * KDIM;
#pragma unroll
            for (int it = 0; it < 5; ++it) r[it] = wt[goff[it]];
        }

        // Hoist all B fragments: 10x ds_load_2addr_b64, single dscnt drain.
        v2f b[20];
        const float* btc = bt[cur];
#pragma unroll
        for (int kk = 0; kk < 20; ++kk) {
            b[kk].x = btc[n16 * LDS_STRIDE + 4 * kk + koff];
            b[kk].y = btc[n16 * LDS_STRIDE + 4 * kk + koff + 1];
        }

        // Two independent accumulator chains for matrix-pipe ILP.
        v8f acc0 = {0.f, 0.f, 0.f, 0.f, 0.f, 0.f, 0.f, 0.f};
        v8f acc1 = {0.f, 0.f, 0.f, 0.f, 0.f, 0.f, 0.f, 0.f};
#pragma unroll
        for (int kk = 0; kk < 20; kk += 2) {
            acc0 = __builtin_amdgcn_wmma_f32_16x16x4_f32(
                false, a[kk], false, b[kk], (short)0, acc0, false, false);
            acc1 = __builtin_amdgcn_wmma_f32_16x16x4_f32(
                false, a[kk + 1], false, b[kk + 1], (short)0, acc1, false, false);
        }

        const int col = ct * 16 + n16;
        const float s = invn[col];   // argmax of (x.w_j)/||w_j||  ==  argmin dist
#pragma unroll
        for (int v = 0; v < 8; ++v) {
            float sc = (acc0[v] + acc1[v]) * s;
            if (sc > best[v]) { best[v] = sc; bidx[v] = col; }
        }

        if (more) {
#pragma unroll
            for (int it = 0; it < 5; ++it) bt[cur ^ 1][loff[it]] = r[it];
        }
        __syncthreads();   // one barrier per tile (double-buffered)
    }

    // Butterfly argmax across the 16 lanes of each half-wave (tile columns).
#pragma unroll
    for (int m = 1; m < 16; m <<= 1) {
#pragma unroll
        for (int v = 0; v < 8; ++v) {
            float ob = __shfl_xor(best[v], m, 32);
            int   oi = __shfl_xor(bidx[v], m, 32);
            if (ob > best[v] || (ob == best[v] && oi < bidx[v])) { best[v] = ob; bidx[v] = oi; }
        }
    }

    if (n16 == 0) {
        const int rbase = rowTile * 16 + ((lane >= 16) ? 8 : 0);
#pragma unroll
        for (int v = 0; v < 8; ++v) ind[rbase + v] = bidx[v];
    }
}

// ---------------- kernel 3: gather quantized rows + loss reduction ----------------
__global__ __launch_bounds__(256) void vq_gather_loss_kernel(
    const float* __restrict__ X, const float* __restrict__ W,
    const int* __restrict__ ind, float* __restrict__ out,
    float* __restrict__ acc, long long total) {
    long long g = (long long)blockIdx.x * blockDim.x + threadIdx.x;
    float d2 = 0.0f;
    if (g < total) {
        long long i = g / KDIM;
        int d = (int)(g - i * KDIM);
        float q = W[(size_t)ind[i] * KDIM + d];
        float x = X[g];
        out[g] = q;                 // straight-through forward value == q
        float diff = q - x;
        d2 = diff * diff;
    }
#pragma unroll
    for (int m = 16; m > 0; m >>= 1) d2 += __shfl_xor(d2, m, 32);
    __shared__ float sred[8];
    int lane = threadIdx.x & 31, w = threadIdx.x >> 5;
    if (lane == 0) sred[w] = d2;
    __syncthreads();
    if (threadIdx.x < 8) {
        float v = sred[threadIdx.x];
#pragma unroll
        for (int m = 4; m > 0; m >>= 1) v += __shfl_xor(v, m, 32);
        if (threadIdx.x == 0) atomicAdd(acc, v);
    }
}

// ---------------- kernel 4: finalize loss ----------------
__global__ void vq_finalize_kernel(const float* acc, float* loss_out, float inv_total) {
    // loss = q_latent + 0.25 * e_latent = 1.25 * mean(diff^2)
    loss_out[0] = acc[0] * 1.25f * inv_total;
}

extern "C" void kernel_launch(void* const* d_in, const int* in_sizes, int n_in,
                              void* d_out, int out_size, void* d_ws, size_t ws_size,
                              hipStream_t stream) {
    const float* X = (const float*)d_in[0];   // cnt_emb [64,2048,80]
    const float* W = (const float*)d_in[1];   // embed_weight [1024,80]
    float* out = (float*)d_out;               // [N*80 quantized] + [1 loss]

    const int N = in_sizes[0] / KDIM;         // 131072 rows
    const int E = in_sizes[1] / KDIM;         // 1024 embeddings
    const long long total = (long long)N * KDIM;

    float* invn = (float*)d_ws;               // [E]
    int*   ind  = (int*)(invn + E);           // [N]
    float* acc  = (float*)(ind + N);          // [1]

    vq_init_kernel<<<1, 1, 0, stream>>>(acc);
    vq_invnorm_kernel<<<(E + 255) / 256, 256, 0, stream>>>(W, invn, E);

    // one wave per 16-row tile, 8 waves (256 threads) per block
    const int rowTiles = N / 16;              // 8192
    vq_argmax_kernel<<<rowTiles / 8, 256, 0, stream>>>(X, W, invn, ind, E);

    const long long nBlocks = (total + 255) / 256;
    vq_gather_loss_kernel<<<(int)nBlocks, 256, 0, stream>>>(X, W, ind, out, acc, total);

    vq_finalize_kernel<<<1, 1, 0, stream>>>(acc, out + total, 1.0f / (float)total);
}